// QAConv_6674379178782
// MI455X (gfx1250) — compile-verified
//
#include <hip/hip_runtime.h>

typedef float v2f __attribute__((ext_vector_type(2)));
typedef float v8f __attribute__((ext_vector_type(8)));

#define HW   192
#define NC   128
#define PAD  132                 // 128 + 4 floats padding -> conflict-free b64 LDS reads
#define NEG_INF (-3.402823466e38f)
#define NBLK 2048                // g*p = 32*64
#define NTHR 384                 // 12 waves, one per 16-row tile-row

// ---------------------------------------------------------------------------
// Kernel 1: per (g,p) pair, compute S = K_p * F_g^T (192x192) with fp32 WMMA
// tiles, track row-max / col-max on the fly, emit sum & sumsq of the 384 maxes.
// ---------------------------------------------------------------------------
__global__ __launch_bounds__(NTHR) void qaconv_pair_kernel(
    const float* __restrict__ features,   // (32,128,192)
    const float* __restrict__ kern,       // (64,128,192)
    float* __restrict__ sums,             // (2048)
    float* __restrict__ sumsqs)           // (2048)
{
  extern __shared__ float smem[];
  float* Kt      = smem;                  // [192][PAD]  K_p transposed: [yx][c]
  float* Ft      = Kt + HW * PAD;         // [192][PAD]  F_g transposed: [hw][c]
  float* colpart = Ft + HW * PAD;         // [12][192]   per-wave col-max partials
  float* rowmax  = colpart + 12 * HW;     // [192]
  float* red     = rowmax + HW;           // [2]

  const int tid = threadIdx.x;
  const int b   = blockIdx.x;
  const int g   = b >> 6;
  const int p   = b & 63;
  const float* Fsrc = features + (size_t)g * NC * HW;
  const float* Ksrc = kern     + (size_t)p * NC * HW;

  // Stage both panels transposed into LDS (coalesced global reads).
  for (int idx = tid; idx < NC * HW; idx += NTHR) {
    int c  = idx / HW;
    int yx = idx - c * HW;
    Kt[yx * PAD + c] = Ksrc[idx];
    Ft[yx * PAD + c] = Fsrc[idx];
  }
  for (int j = tid; j < 12 * HW; j += NTHR) colpart[j] = NEG_INF;
  if (tid == 0) { red[0] = 0.f; red[1] = 0.f; }
  __syncthreads();

  const int w    = tid >> 5;          // wave id 0..11 == tile-row ty
  const int lane = tid & 31;
  const int n    = lane & 15;         // N (and M-row within half) index
  const int h    = lane >> 4;         // half select
  const int koff = 2 * h;             // K offset per WMMA fragment layout

  const int ty = w;
  float rowacc[8];
#pragma unroll
  for (int r = 0; r < 8; ++r) rowacc[r] = NEG_INF;

  const float* Krow = &Kt[(ty * 16 + n) * PAD + koff];

  for (int tx = 0; tx < 12; ++tx) {
    const float* Frow = &Ft[(tx * 16 + n) * PAD + koff];
    v8f acc = {0.f, 0.f, 0.f, 0.f, 0.f, 0.f, 0.f, 0.f};
#pragma unroll
    for (int kk = 0; kk < 32; ++kk) {
      v2f a  = *(const v2f*)(Krow + 4 * kk);   // A[m][2h], A[m][2h+1]
      v2f bb = *(const v2f*)(Frow + 4 * kk);   // B[2h][n], B[2h+1][n]
      acc = __builtin_amdgcn_wmma_f32_16x16x4_f32(
          /*neg_a=*/false, a, /*neg_b=*/false, bb,
          /*c_mod=*/(short)0, acc, /*reuse_a=*/false, /*reuse_b=*/false);
    }
    // per-tile max bookkeeping
    float cm = NEG_INF;
#pragma unroll
    for (int r = 0; r < 8; ++r) {
      float v = acc[r];
      rowacc[r] = fmaxf(rowacc[r], v);   // max over N for row (ty*16 + r + 8h)
      cm = fmaxf(cm, v);                 // max over this half's 8 rows
    }
    cm = fmaxf(cm, __shfl_xor(cm, 16, 32));  // combine M halves -> col max of tile
    if (h == 0) {
      float* cp = &colpart[w * HW + tx * 16 + n];
      *cp = fmaxf(*cp, cm);              // private per-wave partition: race-free
    }
  }

  // reduce rowacc across the 16 lanes of each half; lane n==0 writes result
#pragma unroll
  for (int r = 0; r < 8; ++r) {
    float v = rowacc[r];
    v = fmaxf(v, __shfl_xor(v, 1, 32));
    v = fmaxf(v, __shfl_xor(v, 2, 32));
    v = fmaxf(v, __shfl_xor(v, 4, 32));
    v = fmaxf(v, __shfl_xor(v, 8, 32));
    if (n == 0) rowmax[ty * 16 + r + 8 * h] = v;
  }
  __syncthreads();

  // finalize col-max across the 12 wave partitions; accumulate sum & sumsq
  float lsum = 0.f, lsq = 0.f;
  for (int j = tid; j < HW; j += NTHR) {
    float cmv = NEG_INF;
#pragma unroll
    for (int ww = 0; ww < 12; ++ww) cmv = fmaxf(cmv, colpart[ww * HW + j]);
    float rv = rowmax[j];
    lsum += cmv + rv;
    lsq  += cmv * cmv + rv * rv;
  }
  atomicAdd(&red[0], lsum);
  atomicAdd(&red[1], lsq);
  __syncthreads();
  if (tid == 0) { sums[b] = red[0]; sumsqs[b] = red[1]; }
}

// ---------------------------------------------------------------------------
// Kernel 2: global BN1 (over all 786432 max values) + uniform-weight FC +
// pair sum + BN2, all exact via linearity. Single workgroup.
// ---------------------------------------------------------------------------
__global__ __launch_bounds__(1024) void qaconv_final_kernel(
    const float* __restrict__ sums, const float* __restrict__ sumsqs,
    const float* __restrict__ bn_w, const float* __restrict__ bn_b,
    const float* __restrict__ fc_w, const float* __restrict__ fc_b,
    const float* __restrict__ lbn_w, const float* __restrict__ lbn_b,
    float* __restrict__ out)   // (p=64, g=32)
{
  __shared__ float sh[NBLK];
  __shared__ float red[4];
  const int tid = threadIdx.x;
  if (tid == 0) { red[0] = 0.f; red[1] = 0.f; red[2] = 0.f; red[3] = 0.f; }
  __syncthreads();

  float s1 = 0.f, s2 = 0.f;
  for (int i = tid; i < NBLK; i += 1024) { s1 += sums[i]; s2 += sumsqs[i]; }
  atomicAdd(&red[0], s1);
  atomicAdd(&red[1], s2);
  __syncthreads();

  const float N    = (float)NBLK * 2.f * (float)HW;   // 786432 elements
  const float m    = red[0] / N;
  const float var  = red[1] / N - m * m;
  const float istd = rsqrtf(var + 1e-5f);
  const float fw0  = fc_w[0];                          // uniform weight 1/192
  const float Wsum = (float)HW * fw0;
  const float a    = istd * bn_w[0];
  const float cst  = 2.f * (bn_b[0] * Wsum + fc_b[0]) - 2.f * a * m * Wsum;

  float ls = 0.f, lq = 0.f;
  for (int i = tid; i < NBLK; i += 1024) {
    float s = a * (sums[i] * fw0) + cst;   // summed two-direction score
    sh[i] = s;
    ls += s;
    lq += s * s;
  }
  atomicAdd(&red[2], ls);
  atomicAdd(&red[3], lq);
  __syncthreads();

  const float m2    = red[2] / (float)NBLK;
  const float v2    = red[3] / (float)NBLK - m2 * m2;
  const float istd2 = rsqrtf(v2 + 1e-5f);
  const float lw = lbn_w[0], lb = lbn_b[0];
  for (int i = tid; i < NBLK; i += 1024) {
    int gg = i >> 6;           // i = g*64 + p
    int pp = i & 63;
    out[pp * 32 + gg] = (sh[i] - m2) * istd2 * lw + lb;
  }
}

extern "C" void kernel_launch(void* const* d_in, const int* in_sizes, int n_in,
                              void* d_out, int out_size, void* d_ws, size_t ws_size,
                              hipStream_t stream) {
  (void)in_sizes; (void)n_in; (void)out_size; (void)ws_size;
  const float* features = (const float*)d_in[0];
  const float* kern     = (const float*)d_in[1];
  const float* bn_w     = (const float*)d_in[2];
  const float* bn_b     = (const float*)d_in[3];
  const float* fc_w     = (const float*)d_in[4];
  const float* fc_b     = (const float*)d_in[5];
  const float* lbn_w    = (const float*)d_in[6];
  const float* lbn_b    = (const float*)d_in[7];

  float* sums   = (float*)d_ws;
  float* sumsqs = sums + NBLK;

  const size_t smem_bytes =
      (size_t)(2 * HW * PAD + 12 * HW + HW + 2) * sizeof(float);  // ~212 KB of 320 KB WGP LDS

  // Allow >default dynamic LDS (WGP has 320 KB); idempotent, capture-safe.
  hipFuncSetAttribute((const void*)qaconv_pair_kernel,
                      hipFuncAttributeMaxDynamicSharedMemorySize,
                      (int)smem_bytes);

  qaconv_pair_kernel<<<NBLK, NTHR, smem_bytes, stream>>>(features, kern, sums, sumsqs);
  qaconv_final_kernel<<<1, 1024, 0, stream>>>(sums, sumsqs, bn_w, bn_b,
                                              fc_w, fc_b, lbn_w, lbn_b,
                                              (float*)d_out);
}